// Generator_26895085208119
// MI455X (gfx1250) — compile-verified
//
#include <hip/hip_runtime.h>

// out[b, w] = cos(x[b, w] + thetas[w])   -- analytic closed form of the
// 16-qubit RX product circuit + <Z_w> readout (product state => exact).
//
// The theta broadcast is fused into a V_WMMA_F32_16X16X4_F32:
//   A (16x4) = 0.25 everywhere, B (4x16) = thetas replicated per row,
//   C = x tile  =>  D[m,n] = x[m,n] + thetas[n].

typedef __attribute__((ext_vector_type(2))) float v2f;
typedef __attribute__((ext_vector_type(8))) float v8f;

__global__ __launch_bounds__(32) void qexpz_wmma_kernel(
    const float* __restrict__ x,       // [256, 16] row-major
    const float* __restrict__ thetas,  // [16]
    float* __restrict__ out)           // [256, 16] row-major
{
  const int lane = threadIdx.x;   // 0..31, full wave32 active
  const int col  = lane & 15;     // N index (wire w)
  const int half = lane >> 4;     // 0: rows 0..7, 1: rows 8..15 of tile
  const int m0   = blockIdx.x * 16;  // batch-tile base row (16 tiles for BSZ=256)

  // A-matrix 16x4: every element 0.25 -> sub-layout irrelevant.
  v2f a; a.x = 0.25f; a.y = 0.25f;

  // B-matrix 4x16: every row = thetas, column n <-> lane n&15.
  const float th = thetas[col];
  v2f b; b.x = th; b.y = th;

  // C-matrix = x tile, documented 16x16 f32 C/D layout:
  // VGPR r holds row (r + 8*half), lane&15 is the column.
  v8f c;
#pragma unroll
  for (int r = 0; r < 8; ++r)
    c[r] = x[(m0 + r + 8 * half) * 16 + col];

  // D = A*B + C  =>  D[m,n] = x[m,n] + thetas[n]
  v8f d = __builtin_amdgcn_wmma_f32_16x16x4_f32(
      /*neg_a=*/false, a, /*neg_b=*/false, b,
      /*c_mod=*/(short)0, c, /*reuse_a=*/false, /*reuse_b=*/false);

  // <Z_w> = cos(phi); inputs are ~N(0,2) so plain cosf is exact enough.
#pragma unroll
  for (int r = 0; r < 8; ++r)
    out[(m0 + r + 8 * half) * 16 + col] = __builtin_cosf(d[r]);
}

extern "C" void kernel_launch(void* const* d_in, const int* in_sizes, int n_in,
                              void* d_out, int out_size, void* d_ws, size_t ws_size,
                              hipStream_t stream) {
  const float* x      = (const float*)d_in[0];  // [256*16]
  const float* thetas = (const float*)d_in[1];  // [16]
  float* out = (float*)d_out;                   // [256*16]
  // 16 tiles of 16x16 over the batch; one wave32 per tile.
  qexpz_wmma_kernel<<<16, 32, 0, stream>>>(x, thetas, out);
}